// Model_63058709840619
// MI455X (gfx1250) — compile-verified
//
#include <hip/hip_runtime.h>

// ---------------------------------------------------------------------------
// Types for CDNA5 WMMA f32 16x16x4:
//   A (16x4 f32)  = 2 VGPRs/lane, B (4x16 f32) = 2 VGPRs/lane, C/D = 8 VGPRs
// ---------------------------------------------------------------------------
typedef float v2f __attribute__((ext_vector_type(2)));
typedef float v8f __attribute__((ext_vector_type(8)));

#define GF_BIAS  1
#define GF_ACCUM 2
#define GF_RELU  4

// C[M,Np](ldc) (+)= A[M,K](lda) @ W[K,Np](ldw) (+ bias) (ReLU)
//
// Block = 128 threads (4 waves). Block computes a 128-row x 16-col strip:
// each wave owns two 16x16 M-tiles of the same N-tile. The K x 16 weight
// slice is staged once per block into LDS (<= 64 KB for K <= 1024), swizzled
// so a lane's B fragment {W[ka][n], W[ka+1][n]} is one 8B-aligned ds_load_b64:
//   lds[(k>>1)*32 + n*2 + (k&1)]
// Out-of-range M sub-tiles clamp A-row pointers to row (lane&15) and skip the
// store, so all waves execute the single __syncthreads().
__global__ __launch_bounds__(128)
void gemm16_wmma_f32(const float* __restrict__ A, int lda,
                     const float* __restrict__ W, int ldw,
                     const float* __restrict__ bias,
                     float* __restrict__ C, int ldc,
                     int M, int K, int flags)
{
    __shared__ float lds[1024 * 16];            // K <= 1024

    const int tid  = threadIdx.x;
    const int wave = tid >> 5;
    const int lane = tid & 31;
    const int half = lane >> 4;
    const int lm   = lane & 15;
    const int tn   = blockIdx.x * 16;

    // ---- cooperative stage of W[:,tn:tn+16] into swizzled LDS ----
    for (int idx = tid; idx < K * 16; idx += 128) {
        const int k = idx >> 4, n = idx & 15;
        lds[(k >> 1) * 32 + n * 2 + (k & 1)] = W[(size_t)k * (size_t)ldw + tn + n];
    }
    __syncthreads();

    const int tm0 = (blockIdx.y * 8 + 2 * wave) * 16;
    const int tm1 = tm0 + 16;
    const bool val0 = tm0 < M;
    const bool val1 = tm1 < M;

    const float* __restrict__ arow0 = A + (size_t)(val0 ? tm0 + lm : lm) * (size_t)lda;
    const float* __restrict__ arow1 = A + (size_t)(val1 ? tm1 + lm : lm) * (size_t)lda;

    v8f acc0 = {0.f, 0.f, 0.f, 0.f, 0.f, 0.f, 0.f, 0.f};
    v8f acc1 = {0.f, 0.f, 0.f, 0.f, 0.f, 0.f, 0.f, 0.f};

    #pragma unroll 4
    for (int k0 = 0; k0 < K; k0 += 4) {
        const int ka = k0 + 2 * half;
        const v2f b  = *(const v2f*)&lds[((k0 >> 1) + half) * 32 + lm * 2];
        const v2f a0 = *(const v2f*)(arow0 + ka);   // 8B aligned: lda even, ka even
        const v2f a1 = *(const v2f*)(arow1 + ka);
        acc0 = __builtin_amdgcn_wmma_f32_16x16x4_f32(false, a0, false, b,
                                                     (short)0, acc0, false, false);
        acc1 = __builtin_amdgcn_wmma_f32_16x16x4_f32(false, a1, false, b,
                                                     (short)0, acc1, false, false);
    }

    const float bv = (flags & GF_BIAS) ? bias[tn + lm] : 0.0f;
    if (val0) {
        #pragma unroll
        for (int g = 0; g < 8; ++g) {
            const size_t idx = (size_t)(tm0 + 8 * half + g) * (size_t)ldc + tn + lm;
            float v = acc0[g] + bv;
            if (flags & GF_ACCUM) v += C[idx];
            if (flags & GF_RELU)  v = fmaxf(v, 0.0f);
            C[idx] = v;
        }
    }
    if (val1) {
        #pragma unroll
        for (int g = 0; g < 8; ++g) {
            const size_t idx = (size_t)(tm1 + 8 * half + g) * (size_t)ldc + tn + lm;
            float v = acc1[g] + bv;
            if (flags & GF_ACCUM) v += C[idx];
            if (flags & GF_RELU)  v = fmaxf(v, 0.0f);
            C[idx] = v;
        }
    }
}

// ---------------------------------------------------------------------------
// Mean-aggregation (PyG-style): scatter-add features + degree, then divide.
// Feature rows are padded to 160 floats (pad cols are exact zeros everywhere).
// ---------------------------------------------------------------------------
__global__ void scatter_add_feat(const float* __restrict__ xsrc,
                                 const int* __restrict__ sidx,
                                 const int* __restrict__ didx,
                                 float* __restrict__ agg, int E)
{
    int tid = blockIdx.x * blockDim.x + threadIdx.x;
    int total = E * 160;
    if (tid >= total) return;
    int e = tid / 160, f = tid - e * 160;
    atomicAdd(&agg[(size_t)didx[e] * 160 + f], xsrc[(size_t)sidx[e] * 160 + f]);
}

__global__ void scatter_count(const int* __restrict__ didx, float* __restrict__ cnt, int E)
{
    int tid = blockIdx.x * blockDim.x + threadIdx.x;
    if (tid < E) atomicAdd(&cnt[didx[tid]], 1.0f);
}

__global__ void div_by_count(float* __restrict__ agg, const float* __restrict__ cnt, int n)
{
    int tid = blockIdx.x * blockDim.x + threadIdx.x;
    if (tid < n) agg[tid] = agg[tid] / fmaxf(cnt[tid / 160], 1.0f);
}

__global__ void relu_inplace(float* __restrict__ x, int n)
{
    int tid = blockIdx.x * blockDim.x + threadIdx.x;
    if (tid < n) x[tid] = fmaxf(x[tid], 0.0f);
}

// ---------------------------------------------------------------------------
// Zero-padded weight / bias packing: [K,N] row-major -> [Kp,Np] row-major
// ---------------------------------------------------------------------------
__global__ void pack_weight(const float* __restrict__ in, float* __restrict__ out,
                            int K, int N, int Kp, int Np)
{
    int tid = blockIdx.x * blockDim.x + threadIdx.x;
    if (tid >= Kp * Np) return;
    int r = tid / Np, c = tid - r * Np;
    out[tid] = (r < K && c < N) ? in[(size_t)r * N + c] : 0.0f;
}

__global__ void pack_bias(const float* __restrict__ in, float* __restrict__ out, int N, int Np)
{
    int tid = blockIdx.x * blockDim.x + threadIdx.x;
    if (tid < Np) out[tid] = (tid < N) ? in[tid] : 0.0f;
}

// ---------------------------------------------------------------------------
// Edge head: t1 = relu(P1[src] + P2[dst])   (hb1 folded into P1)
// ---------------------------------------------------------------------------
__global__ void head_combine(const float* __restrict__ P1, const float* __restrict__ P2,
                             const int* __restrict__ i0, const int* __restrict__ i1,
                             float* __restrict__ t1, int base, int cnt)
{
    int tid = blockIdx.x * blockDim.x + threadIdx.x;
    int total = cnt * 160;
    if (tid >= total) return;
    int r = tid / 160, f = tid - r * 160;
    int a = i0[base + r], b = i1[base + r];
    t1[tid] = fmaxf(P1[(size_t)a * 160 + f] + P2[(size_t)b * 160 + f], 0.0f);
}

__global__ void head_writeout(const float* __restrict__ t3, float* __restrict__ out,
                              int base, int cnt)
{
    int tid = blockIdx.x * blockDim.x + threadIdx.x;
    if (tid >= cnt * 3) return;
    int r = tid / 3, c = tid - r * 3;
    out[(size_t)(base + r) * 3 + c] = t3[(size_t)r * 16 + c];
}

// ---------------------------------------------------------------------------
// Host orchestration
// ---------------------------------------------------------------------------
static inline int cdiv(long long a, long long b) { return (int)((a + b - 1) / b); }

static void launch_gemm(const float* A, int lda, const float* W, int ldw,
                        const float* bias, float* C, int ldc,
                        int M, int Np, int K, int flags, hipStream_t s)
{
    dim3 grid(Np / 16, cdiv(M, 128));           // 4 waves x 2 M-tiles = 128 rows/block
    gemm16_wmma_f32<<<grid, 128, 0, s>>>(A, lda, W, ldw, bias, C, ldc, M, K, flags);
}

extern "C" void kernel_launch(void* const* d_in, const int* in_sizes, int n_in,
                              void* d_out, int out_size, void* d_ws, size_t ws_size,
                              hipStream_t stream)
{
    const int HP = 160;          // H=150 padded
    const int N_LNC = in_sizes[0] / 768;
    const int N_MIR = in_sizes[1] / 768;
    const int N_PRO = in_sizes[2] / 1024;
    const int E_LPI = in_sizes[24] / 2;
    const int E_LMI = in_sizes[25] / 2;
    const int E_MPI = in_sizes[26] / 2;
    const int E_LBL = in_sizes[27] / 2;

    const float* x_lnc = (const float*)d_in[0];
    const float* x_mir = (const float*)d_in[1];
    const float* x_pro = (const float*)d_in[2];
    const int* lpi = (const int*)d_in[24];  const int* lpi_l = lpi, *lpi_p = lpi + E_LPI;
    const int* lmi = (const int*)d_in[25];  const int* lmi_l = lmi, *lmi_m = lmi + E_LMI;
    const int* mpi = (const int*)d_in[26];  const int* mpi_m = mpi, *mpi_p = mpi + E_MPI;
    const int* eli = (const int*)d_in[27];  const int* eli_l = eli, *eli_p = eli + E_LBL;

    // ---- workspace bump allocator (floats) ----
    float* base = (float*)d_ws;
    size_t off = 0;
    auto alloc = [&](size_t n) { float* r = base + off; off += n; return r; };

    float* xlA = alloc((size_t)N_LNC * HP);
    float* xmA = alloc((size_t)N_MIR * HP);
    float* xpA = alloc((size_t)N_PRO * HP);
    float* xlB = alloc((size_t)N_LNC * HP);
    float* xmB = alloc((size_t)N_MIR * HP);
    float* xpB = alloc((size_t)N_PRO * HP);
    float* agg = alloc((size_t)N_LNC * HP);      // largest dst set
    float* cnt = alloc((size_t)N_LNC);
    float* h1  = alloc((size_t)N_LNC * 304);
    float* h2  = alloc((size_t)N_LNC * 208);
    float* h3  = alloc((size_t)N_LNC * HP);
    float* P1  = alloc((size_t)N_LNC * HP);
    float* P2  = alloc((size_t)N_PRO * HP);
    // packed weights / biases
    float* Wle = alloc(768 * HP);   float* ble = alloc(HP);
    float* Wme = alloc(768 * HP);   float* bme = alloc(HP);
    float* Wpe = alloc(1024 * HP);  float* bpe = alloc(HP);
    float* Wlp = alloc((size_t)18 * HP * HP);
    float* Wrp = alloc((size_t)18 * HP * HP);
    float* blp = alloc((size_t)18 * HP);
    float* cW1p = alloc(768 * 304); float* cb1p = alloc(304);
    float* cW2p = alloc(304 * 208); float* cb2p = alloc(208);
    float* cW3p = alloc(208 * HP);  float* cb3p = alloc(HP);
    float* hW1t = alloc(HP * HP);   float* hW1b = alloc(HP * HP);
    float* hb1p = alloc(HP);
    float* hW2p = alloc(HP * 64);   float* hb2p = alloc(64);
    float* hW3p = alloc(64 * 16);   float* hb3p = alloc(16);
    (void)ws_size;

    const int B = 256;
    auto pw = [&](const float* in, float* out, int K, int N, int Kp, int Np) {
        pack_weight<<<cdiv((long long)Kp * Np, B), B, 0, stream>>>(in, out, K, N, Kp, Np);
    };
    auto pb = [&](const float* in, float* out, int N, int Np) {
        pack_bias<<<cdiv(Np, B), B, 0, stream>>>(in, out, N, Np);
    };

    // ---- pack all weights into zero-padded layouts ----
    pw((const float*)d_in[3], Wle, 768, 150, 768, HP);   pb((const float*)d_in[4], ble, 150, HP);
    pw((const float*)d_in[5], Wme, 768, 150, 768, HP);   pb((const float*)d_in[6], bme, 150, HP);
    pw((const float*)d_in[7], Wpe, 1024, 150, 1024, HP); pb((const float*)d_in[8], bpe, 150, HP);
    const float* Wl_raw = (const float*)d_in[9];
    const float* bl_raw = (const float*)d_in[10];
    const float* Wr_raw = (const float*)d_in[11];
    for (int i = 0; i < 18; ++i) {
        pw(Wl_raw + (size_t)i * 150 * 150, Wlp + (size_t)i * HP * HP, 150, 150, HP, HP);
        pw(Wr_raw + (size_t)i * 150 * 150, Wrp + (size_t)i * HP * HP, 150, 150, HP, HP);
        pb(bl_raw + (size_t)i * 150, blp + (size_t)i * HP, 150, HP);
    }
    pw((const float*)d_in[12], cW1p, 768, 300, 768, 304); pb((const float*)d_in[13], cb1p, 300, 304);
    pw((const float*)d_in[14], cW2p, 300, 200, 304, 208); pb((const float*)d_in[15], cb2p, 200, 208);
    pw((const float*)d_in[16], cW3p, 200, 150, 208, HP);  pb((const float*)d_in[17], cb3p, 150, HP);
    const float* hW1 = (const float*)d_in[18];
    pw(hW1, hW1t, 150, 150, HP, HP);                       // rows 0..149  (h part)
    pw(hW1 + 150 * 150, hW1b, 150, 150, HP, HP);           // rows 150..299 (xp3 part)
    pb((const float*)d_in[19], hb1p, 150, HP);
    pw((const float*)d_in[20], hW2p, 150, 50, HP, 64);     pb((const float*)d_in[21], hb2p, 50, 64);
    pw((const float*)d_in[22], hW3p, 50, 3, 64, 16);       pb((const float*)d_in[23], hb3p, 3, 16);

    // ---- node-type input projections (fp32 WMMA GEMMs) ----
    launch_gemm(x_lnc, 768,  Wle, HP, ble, xlA, HP, N_LNC, HP, 768,  GF_BIAS, stream);
    launch_gemm(x_mir, 768,  Wme, HP, bme, xmA, HP, N_MIR, HP, 768,  GF_BIAS, stream);
    launch_gemm(x_pro, 1024, Wpe, HP, bpe, xpA, HP, N_PRO, HP, 1024, GF_BIAS, stream);

    // ---- one SAGE conv: out (+)= mean_aggr(xsrc)@Wl[l,et] + bl[l,et] + xdst@Wr[l,et]
    auto conv = [&](int l, int et, const float* xsrc, const int* sidx, const int* didx,
                    int E, const float* xdst, int Ndst, float* out, bool first) {
        hipMemsetAsync(agg, 0, (size_t)Ndst * HP * sizeof(float), stream);
        hipMemsetAsync(cnt, 0, (size_t)Ndst * sizeof(float), stream);
        scatter_add_feat<<<cdiv((long long)E * HP, B), B, 0, stream>>>(xsrc, sidx, didx, agg, E);
        scatter_count<<<cdiv(E, B), B, 0, stream>>>(didx, cnt, E);
        div_by_count<<<cdiv((long long)Ndst * HP, B), B, 0, stream>>>(agg, cnt, Ndst * HP);
        const int wi = l * 6 + et;
        launch_gemm(agg,  HP, Wlp + (size_t)wi * HP * HP, HP, blp + (size_t)wi * HP,
                    out, HP, Ndst, HP, HP, GF_BIAS | (first ? 0 : GF_ACCUM), stream);
        launch_gemm(xdst, HP, Wrp + (size_t)wi * HP * HP, HP, nullptr,
                    out, HP, Ndst, HP, HP, GF_ACCUM, stream);
    };

    float *cl = xlA, *cm = xmA, *cp = xpA;
    float *nl = xlB, *nm = xmB, *np_ = xpB;
    for (int l = 0; l < 3; ++l) {
        if (l < 2) {
            conv(l, 1, cp, lpi_p, lpi_l, E_LPI, cl, N_LNC, nl, true);
            conv(l, 3, cm, lmi_m, lmi_l, E_LMI, cl, N_LNC, nl, false);
            conv(l, 2, cl, lmi_l, lmi_m, E_LMI, cm, N_MIR, nm, true);
            conv(l, 5, cp, mpi_p, mpi_m, E_MPI, cm, N_MIR, nm, false);
        }
        conv(l, 0, cl, lpi_l, lpi_p, E_LPI, cp, N_PRO, np_, true);
        conv(l, 4, cm, mpi_m, mpi_p, E_MPI, cp, N_PRO, np_, false);
        if (l < 2) {
            relu_inplace<<<cdiv((long long)N_LNC * HP, B), B, 0, stream>>>(nl,  N_LNC * HP);
            relu_inplace<<<cdiv((long long)N_MIR * HP, B), B, 0, stream>>>(nm,  N_MIR * HP);
            relu_inplace<<<cdiv((long long)N_PRO * HP, B), B, 0, stream>>>(np_, N_PRO * HP);
        }
        float* t;
        t = cl; cl = nl; nl = t;
        t = cm; cm = nm; nm = t;
        t = cp; cp = np_; np_ = t;
    }
    const float* xp3 = cp;   // layer-3 protein embeddings

    // ---- classifier MLP on raw lncRNA features ----
    launch_gemm(x_lnc, 768, cW1p, 304, cb1p, h1, 304, N_LNC, 304, 768, GF_BIAS | GF_RELU, stream);
    launch_gemm(h1,    304, cW2p, 208, cb2p, h2, 208, N_LNC, 208, 304, GF_BIAS | GF_RELU, stream);
    launch_gemm(h2,    208, cW3p, HP,  cb3p, h3, HP,  N_LNC, HP,  208, GF_BIAS, stream);

    // ---- head layer 1, restructured: ef@hW1+hb1 == h@hW1t + xp3@hW1b + hb1 ----
    launch_gemm(h3,  HP, hW1t, HP, hb1p,   P1, HP, N_LNC, HP, HP, GF_BIAS, stream);
    launch_gemm(xp3, HP, hW1b, HP, nullptr, P2, HP, N_PRO, HP, HP, 0, stream);

    // ---- edge head, chunked; reuse dead classifier buffers as scratch ----
    const int CH = 65536;                       // multiple of 16; tail (41248) also is
    float* t1 = h1;                             // CH*160 <= N_LNC*304
    float* t2 = h2;                             // CH*64  <= N_LNC*208
    float* t3 = h2 + (size_t)CH * 80;           // CH*16, disjoint from t2 (CH*64)
    float* out = (float*)d_out;
    for (int b0 = 0; b0 < E_LBL; b0 += CH) {
        const int c = (E_LBL - b0 < CH) ? (E_LBL - b0) : CH;
        head_combine<<<cdiv((long long)c * HP, B), B, 0, stream>>>(P1, P2, eli_l, eli_p, t1, b0, c);
        launch_gemm(t1, HP, hW2p, 64, hb2p, t2, 64, c, 64, HP, GF_BIAS | GF_RELU, stream);
        launch_gemm(t2, 64, hW3p, 16, hb3p, t3, 16, c, 16, 64, GF_BIAS, stream);
        head_writeout<<<cdiv((long long)c * 3, B), B, 0, stream>>>(t3, out, b0, c);
    }
}